// ResBlock_59141699666450
// MI455X (gfx1250) — compile-verified
//
#include <hip/hip_runtime.h>

#define NN   100000
#define NE   1600000
#define CIN  64
#define COUT 128
#define EPSF 1e-5f

typedef float v2f __attribute__((ext_vector_type(2)));
typedef float v8f __attribute__((ext_vector_type(8)));

// ---------------------------------------------------------------- zero
__global__ __launch_bounds__(256) void zero_f32_kernel(float* __restrict__ p, int n4) {
    int i = blockIdx.x * 256 + threadIdx.x;
    if (i < n4) {
        float4 z = {0.f, 0.f, 0.f, 0.f};
        ((float4*)p)[i] = z;
    }
}

// ------------------------------------------------- edge scatter-add (agg)
template<int K>
__global__ __launch_bounds__(256) void scatter_add_kernel(
    const float* __restrict__ feat, const int* __restrict__ src,
    const int* __restrict__ dst, float* __restrict__ agg) {
    constexpr int CH = K / 4;                 // float4 chunks per edge
    int tid = blockIdx.x * 256 + threadIdx.x;
    int e = tid / CH;                         // CH is power of two -> shift
    if (e >= NE) return;
    int c4 = (tid % CH) * 4;
    int s = src[e], d = dst[e];
    float4 v = *(const float4*)(feat + s * K + c4);
    float* o = agg + d * K + c4;
    atomicAdd(o + 0, v.x);
    atomicAdd(o + 1, v.y);
    atomicAdd(o + 2, v.z);
    atomicAdd(o + 3, v.w);
}

// ------------------------------------- fused dual GEMM: A1*W1^T + A2*W2^T + b
// block = 256 threads = 8 waves; block tile = 16 rows x 128 cols; wave tile 16x16
template<int K>
__global__ __launch_bounds__(256) void graphconv_wmma_kernel(
    const float* __restrict__ A1, const float* __restrict__ A2,
    const float* __restrict__ W1, const float* __restrict__ W2,
    const float* __restrict__ bias, float* __restrict__ out) {
    constexpr int LDK = K + 2;                 // pad: conflict-free ds_load_b64
    __shared__ float sA1[16 * LDK];
    __shared__ float sA2[16 * LDK];
    const int tid  = threadIdx.x;
    const int row0 = blockIdx.x << 4;
    for (int i = tid; i < 16 * K; i += 256) {  // coalesced cooperative stage
        int r = i / K, c = i % K;
        sA1[r * LDK + c] = A1[(row0 + r) * K + c];
        sA2[r * LDK + c] = A2[(row0 + r) * K + c];
    }
    __syncthreads();

    const int lane = tid & 31;
    const int wave = tid >> 5;
    const int mn   = lane & 15;     // M for A-frag, N for B/C-frag
    const int half = lane >> 4;     // K-pair select per ISA layout
    const int col0 = wave << 4;
    const int koff = half << 1;

    v8f acc = {0.f, 0.f, 0.f, 0.f, 0.f, 0.f, 0.f, 0.f};
    const float* w1r = W1 + (col0 + mn) * K;
    const float* w2r = W2 + (col0 + mn) * K;
#pragma unroll
    for (int k = 0; k < K; k += 4) {
        v2f a = *(const v2f*)(sA1 + mn * LDK + k + koff);
        v2f b = *(const v2f*)(w1r + k + koff);
        acc = __builtin_amdgcn_wmma_f32_16x16x4_f32(false, a, false, b,
                                                    (short)0, acc, false, false);
    }
#pragma unroll
    for (int k = 0; k < K; k += 4) {
        v2f a = *(const v2f*)(sA2 + mn * LDK + k + koff);
        v2f b = *(const v2f*)(w2r + k + koff);
        acc = __builtin_amdgcn_wmma_f32_16x16x4_f32(false, a, false, b,
                                                    (short)0, acc, false, false);
    }
    const float bb    = bias[col0 + mn];
    const int   rbase = row0 + (half << 3);    // C layout: lanes>=16 hold M+8
#pragma unroll
    for (int i = 0; i < 8; i++)
        out[(rbase + i) * COUT + col0 + mn] = acc[i] + bb;
}

// ----------------------------------------------- per-column sum / sum-of-squares
__global__ __launch_bounds__(256) void bn_stats_kernel(
    const float* __restrict__ h, float* __restrict__ sum, float* __restrict__ sumsq) {
    __shared__ float rs[256], rq[256];
    const int tid  = threadIdx.x;
    const int c    = tid & 127;
    const int sg   = tid >> 7;
    const int r0   = blockIdx.x << 8;
    const int rend = (r0 + 256 < NN) ? (r0 + 256) : NN;
    float s = 0.f, q = 0.f;
    for (int r = r0 + sg; r < rend; r += 2) {
        float v = h[r * COUT + c];
        s += v; q += v * v;
    }
    rs[tid] = s; rq[tid] = q;
    __syncthreads();
    if (tid < 128) {
        atomicAdd(&sum[tid],   rs[tid] + rs[tid + 128]);
        atomicAdd(&sumsq[tid], rq[tid] + rq[tid + 128]);
    }
}

// ----------------------------------------------- apply BN + ReLU, in place
__global__ __launch_bounds__(256) void bn_relu_kernel(
    float* __restrict__ h, const float* __restrict__ sum,
    const float* __restrict__ sumsq, const float* __restrict__ g,
    const float* __restrict__ be) {
    int i = blockIdx.x * 256 + threadIdx.x;
    if (i >= NN * COUT / 4) return;
    int c = (i << 2) & 127;
    float4 v = ((const float4*)h)[i];
    float o[4] = {v.x, v.y, v.z, v.w};
    const float invN = 1.f / (float)NN;
#pragma unroll
    for (int j = 0; j < 4; j++) {
        float mu  = sum[c + j] * invN;
        float var = sumsq[c + j] * invN - mu * mu;
        float sc  = g[c + j] * rsqrtf(var + EPSF);
        float t   = (o[j] - mu) * sc + be[c + j];
        o[j] = t > 0.f ? t : 0.f;
    }
    float4 w = {o[0], o[1], o[2], o[3]};
    ((float4*)h)[i] = w;
}

// ---------------- final: lin = x*Wlin^T (WMMA) ; out = relu(bn1(h1)+lin+blin)
// io holds h1 on entry (in d_out), overwritten with final result.
__global__ __launch_bounds__(256) void final_wmma_kernel(
    const float* __restrict__ x, const float* __restrict__ Wlin,
    const float* __restrict__ blin, float* __restrict__ io,
    const float* __restrict__ sum1, const float* __restrict__ sumsq1,
    const float* __restrict__ g1, const float* __restrict__ be1) {
    constexpr int K = CIN;
    constexpr int LDK = K + 2;
    __shared__ float sA[16 * LDK];
    const int tid  = threadIdx.x;
    const int row0 = blockIdx.x << 4;
    for (int i = tid; i < 16 * K; i += 256) {
        int r = i / K, c = i % K;
        sA[r * LDK + c] = x[(row0 + r) * K + c];
    }
    __syncthreads();

    const int lane = tid & 31;
    const int wave = tid >> 5;
    const int mn   = lane & 15;
    const int half = lane >> 4;
    const int col0 = wave << 4;
    const int koff = half << 1;

    v8f acc = {0.f, 0.f, 0.f, 0.f, 0.f, 0.f, 0.f, 0.f};
    const float* wr = Wlin + (col0 + mn) * K;
#pragma unroll
    for (int k = 0; k < K; k += 4) {
        v2f a = *(const v2f*)(sA + mn * LDK + k + koff);
        v2f b = *(const v2f*)(wr + k + koff);
        acc = __builtin_amdgcn_wmma_f32_16x16x4_f32(false, a, false, b,
                                                    (short)0, acc, false, false);
    }
    const int   col  = col0 + mn;
    const float invN = 1.f / (float)NN;
    const float mu   = sum1[col] * invN;
    const float var  = sumsq1[col] * invN - mu * mu;
    const float sc   = g1[col] * rsqrtf(var + EPSF);
    const float beta = be1[col];
    const float bl   = blin[col];
    const int rbase  = row0 + (half << 3);
#pragma unroll
    for (int i = 0; i < 8; i++) {
        int   idx = (rbase + i) * COUT + col;
        float bn  = (io[idx] - mu) * sc + beta;
        float o   = bn + acc[i] + bl;
        io[idx] = o > 0.f ? o : 0.f;
    }
}

// ----------------------------------------------------------------------------
extern "C" void kernel_launch(void* const* d_in, const int* in_sizes, int n_in,
                              void* d_out, int out_size, void* d_ws, size_t ws_size,
                              hipStream_t stream) {
    (void)in_sizes; (void)n_in; (void)out_size; (void)ws_size;
    const float* x    = (const float*)d_in[0];
    const int*   ei   = (const int*)d_in[1];
    const int*   src  = ei;            // edge_index[0]
    const int*   dst  = ei + NE;       // edge_index[1]
    const float* Wr0  = (const float*)d_in[2];
    const float* Wn0  = (const float*)d_in[3];
    const float* b0   = (const float*)d_in[4];
    const float* g0   = (const float*)d_in[5];
    const float* be0  = (const float*)d_in[6];
    const float* Wr1  = (const float*)d_in[7];
    const float* Wn1  = (const float*)d_in[8];
    const float* b1   = (const float*)d_in[9];
    const float* g1   = (const float*)d_in[10];
    const float* be1  = (const float*)d_in[11];
    const float* Wlin = (const float*)d_in[12];
    const float* blin = (const float*)d_in[13];
    float* out = (float*)d_out;
    float* ws  = (float*)d_ws;

    float* sum0   = ws;                     // 128
    float* sumsq0 = ws + 128;               // 128
    float* sum1   = ws + 256;               // 128
    float* sumsq1 = ws + 384;               // 128
    float* agg0   = ws + 512;               // N*64
    float* agg1   = agg0 + NN * CIN;        // N*128
    float* h0     = agg1 + NN * COUT;       // N*128 (becomes y in place)

    // zero stats + agg buffers (accumulated into every call)
    const int zeroN = 512 + NN * (CIN + COUT);     // divisible by 4
    zero_f32_kernel<<<(zeroN / 4 + 255) / 256, 256, 0, stream>>>(ws, zeroN / 4);

    // conv0 aggregation + GEMM + BN0 + ReLU
    scatter_add_kernel<CIN><<<NE * (CIN / 4) / 256, 256, 0, stream>>>(x, src, dst, agg0);
    graphconv_wmma_kernel<CIN><<<NN / 16, 256, 0, stream>>>(x, agg0, Wr0, Wn0, b0, h0);
    bn_stats_kernel<<<(NN + 255) / 256, 256, 0, stream>>>(h0, sum0, sumsq0);
    bn_relu_kernel<<<NN * COUT / 4 / 256, 256, 0, stream>>>(h0, sum0, sumsq0, g0, be0);

    // conv1 aggregation + GEMM (h1 -> d_out) + BN1 stats
    scatter_add_kernel<COUT><<<NE * (COUT / 4) / 256, 256, 0, stream>>>(h0, src, dst, agg1);
    graphconv_wmma_kernel<COUT><<<NN / 16, 256, 0, stream>>>(h0, agg1, Wr1, Wn1, b1, out);
    bn_stats_kernel<<<(NN + 255) / 256, 256, 0, stream>>>(out, sum1, sumsq1);

    // residual lin GEMM + BN1 apply + add + ReLU
    final_wmma_kernel<<<NN / 16, 256, 0, stream>>>(x, Wlin, blin, out,
                                                   sum1, sumsq1, g1, be1);
}